// RankingLossListWiseDistil_89249420411005
// MI455X (gfx1250) — compile-verified
//
#include <hip/hip_runtime.h>
#include <hip/hip_bf16.h>
#include <math.h>

// ---------------------------------------------------------------------------
// RankingLossListWiseDistil for MI455X (gfx1250, wave32).
//
// B=64 lists, N=512 items. VALU/transcendental bound; inputs staged in LDS.
// - Per-item data packed in one float4 -> single broadcast ds_load_b128/iter.
// - Validity folded into the label encoding (invalid -> -1e30), no extra LDS.
// - Each list's pairwise loop split over 8 chunk-blocks (512 blocks total)
//   for WGP utilization on a latency-bound problem.
// - All cross-lane reductions via V_WMMA_F32_16X16X4_F32 against all-ones B
//   (f32-exact, deterministic, layout-robust), then fixed-order LDS combine.
// ---------------------------------------------------------------------------

typedef __attribute__((ext_vector_type(2))) float v2f;
typedef __attribute__((ext_vector_type(8))) float v8f;

#define BATCH 64
#define NITEMS 512
#define CHUNKS 8
#define CHUNK (NITEMS / CHUNKS)  // 64 j-columns per block
#define LOG_EPS_F (-23.025850929940457f)
#define INVALID_LAB (-1.0e30f)
#define TOPN 50
#define SMOOTH 0.25f

// Sum the 32 lane values of a wave using one f32 WMMA.
// Lane value -> one unique A slot (A VGPR0 = v, VGPR1 = 0), B = all-ones.
// sum(lane's 8 D VGPRs) + xor-16 partner = sum of all A elements = wave sum,
// independent of exact operand lane layout. EXEC must be all ones here.
__device__ __forceinline__ float wave_sum32_wmma(float v) {
  v2f a;
  a.x = v;
  a.y = 0.0f;
  v2f b;
  b.x = 1.0f;
  b.y = 1.0f;
  v8f c = {};
  c = __builtin_amdgcn_wmma_f32_16x16x4_f32(false, a, false, b, (short)0, c,
                                            false, false);
  float s = c[0] + c[1] + c[2] + c[3] + c[4] + c[5] + c[6] + c[7];
  s += __shfl_xor(s, 16, 32);
  return s;  // full wave sum in every lane
}

__global__ __launch_bounds__(NITEMS) void rank_loss_chunk_kernel(
    const float* __restrict__ logits, const float* __restrict__ labels,
    float* __restrict__ partials) {
  // Per-item packed record: {masked_logit, encoded_label, discount, capped_rank}
  __shared__ float4 sh_pack[NITEMS];
  __shared__ float sh_sl[NITEMS];      // masked logits (rank phase)
  __shared__ float sh_reld[TOPN + 1];  // relative-discount LUT by rank diff
  __shared__ float sh_wsum[NITEMS / 32];
  __shared__ float sh_idcg;

  const int b = blockIdx.x >> 3;        // list index
  const int cch = blockIdx.x & (CHUNKS - 1);
  const int j0 = cch * CHUNK;           // this block's j-column window
  const int i = threadIdx.x;

  // ---- Stage + mask inputs -------------------------------------------------
  const float lab_raw = labels[b * NITEMS + i];
  const float lg_raw = logits[b * NITEMS + i];
  const bool valid = lab_raw > -1000.0f;
  const float lab_enc = valid ? lab_raw : INVALID_LAB;
  const float sl = valid ? lg_raw : LOG_EPS_F;
  sh_sl[i] = sl;

  // reld[d] = |1/log1p(d) - 1/log1p(d+1)|, reld[0] = 0, d in [0,50]
  if (i <= TOPN) {
    sh_reld[i] = (i == 0) ? 0.0f
                          : fabsf(1.0f / logf((float)(i + 1)) -
                                  1.0f / logf((float)(i + 2)));
  }
  __syncthreads();

  // ---- Ranks via stable comparison counting (== stable double argsort) -----
  // Vectorized LDS reads: 128 x ds_load_b128, broadcast across lanes.
  const float li = sl;
  int cnt = 0;
  const float4* sl4 = (const float4*)sh_sl;
#pragma unroll 4
  for (int j4 = 0; j4 < NITEMS / 4; ++j4) {
    const float4 q = sl4[j4];
    const int j = 4 * j4;
    cnt += (int)((q.x > li) || (q.x == li && (j + 0) < i));
    cnt += (int)((q.y > li) || (q.y == li && (j + 1) < i));
    cnt += (int)((q.z > li) || (q.z == li && (j + 2) < i));
    cnt += (int)((q.w > li) || (q.w == li && (j + 3) < i));
  }
  const int rank = cnt + 1;  // 1-based descending rank
  const float disc = (rank > TOPN) ? 0.0f : 1.0f / logf((float)(rank + 1));
  const float capf = (float)((rank > TOPN) ? (TOPN + 1) : rank);
  sh_pack[i] = make_float4(sl, lab_enc, disc, capf);

  // ideal-DCG normalizer: positional (NOT rank-based), first 50 slots only
  if (i == 0) {
    float md = 0.0f;
    for (int n = 0; n < TOPN; ++n) {
      const float v = (labels[b * NITEMS + n] > -1000.0f) ? 1.0f : 0.0f;
      md += v * (1.0f / logf((float)(n + 2)));
    }
    sh_idcg = (md > 0.0f) ? 1.0f / md : 0.0f;
  }
  __syncthreads();

  // ---- Pairwise loop over this block's j-window ---------------------------
  // bce = t*softplus(-x) + (1-t)*softplus(x) = softplus(x) - t*x
  // mask = (pld > 0) && lab_j valid   (invalid-i implied: pld = -huge)
  const float Li = lab_enc;
  const float di = disc;
  const float cif = capf;

  float acc = 0.0f;
#pragma unroll 4
  for (int jj = 0; jj < CHUNK; ++jj) {
    const float4 p = sh_pack[j0 + jj];  // one broadcast ds_load_b128

    const float pld = Li - p.y;
    const float m = (pld > 0.0f && p.y > -1000.0f) ? 1.0f : 0.0f;

    const float x = (li - p.x) * 0.5f;
    const float t = 1.0f / (1.0f + __expf(-pld * 0.5f));  // sigmoid(pld/2)
    const float sp =
        fmaxf(x, 0.0f) + __logf(1.0f + __expf(-fabsf(x)));  // softplus(x)
    const float bce = sp - t * x;

    const float ad = fabsf(di - p.z);
    const float rdf = fabsf(cif - p.w);          // exact small integer
    const float rel = sh_reld[(int)rdf];         // LUT gather
    const float w = (1.0f - SMOOTH) * rel + SMOOTH * ad;

    acc = __builtin_fmaf(bce * w, m, acc);
  }

  // ---- Deterministic reduction: WMMA per wave, LDS across waves -----------
  const float wsum = wave_sum32_wmma(acc);
  const int wave = i >> 5;
  if ((i & 31) == 0) sh_wsum[wave] = wsum;
  __syncthreads();

  if (i == 0) {
    float s = 0.0f;
#pragma unroll
    for (int w = 0; w < NITEMS / 32; ++w) s += sh_wsum[w];
    partials[blockIdx.x] = s * sh_idcg;  // idcg-scaled chunk partial
  }
}

// Sum of BATCH*CHUNKS = 512 idcg-scaled partials / BATCH (mean over lists). Fixed order, WMMA wave
// reductions, single block -> deterministic.
__global__ __launch_bounds__(256) void finalize_mean_kernel(
    const float* __restrict__ partials, float* __restrict__ out) {
  __shared__ float sh_wsum[8];
  const int t = threadIdx.x;  // 0..255
  const float v = partials[t] + partials[t + 256];
  const float s = wave_sum32_wmma(v);
  if ((t & 31) == 0) sh_wsum[t >> 5] = s;
  __syncthreads();
  if (t == 0) {
    float tot = 0.0f;
#pragma unroll
    for (int w = 0; w < 8; ++w) tot += sh_wsum[w];
    out[0] = tot * (1.0f / (float)BATCH);
  }
}

extern "C" void kernel_launch(void* const* d_in, const int* in_sizes, int n_in,
                              void* d_out, int out_size, void* d_ws,
                              size_t ws_size, hipStream_t stream) {
  (void)in_sizes;
  (void)n_in;
  (void)out_size;
  (void)ws_size;
  const float* logits = (const float*)d_in[0];
  const float* labels = (const float*)d_in[1];
  float* out = (float*)d_out;
  float* ws = (float*)d_ws;  // BATCH*CHUNKS idcg-scaled partials

  rank_loss_chunk_kernel<<<BATCH * CHUNKS, NITEMS, 0, stream>>>(logits, labels,
                                                                ws);
  finalize_mean_kernel<<<1, 256, 0, stream>>>(ws, out);
}